// PointNetSetAbstraction_11098195493385
// MI455X (gfx1250) — compile-verified
//
#include <hip/hip_runtime.h>
#include <hip/hip_bf16.h>
#include <cfloat>

#define NPTS   16384
#define BATCH  4
#define SQ     1024
#define NSAMP  32
#define PTOT   (BATCH * NSAMP * SQ)   /* 131072 positions */
#define C1     67
#define O1     64
#define O2     64
#define O3     128
#define RAD2   0.04f
#define EPS_BN 1e-5f
#define QOFF   (BATCH * SQ * 3)       /* 12288 */
#define COLS3  (BATCH * SQ)           /* 4096  */

typedef float v2f __attribute__((ext_vector_type(2)));
typedef float v8f __attribute__((ext_vector_type(8)));

__device__ __forceinline__ v8f wmma4(v2f a, v2f b, v8f c) {
  // D(16x16,f32) = A(16x4,f32) * B(4x16,f32) + C   -> v_wmma_f32_16x16x4_f32
  return __builtin_amdgcn_wmma_f32_16x16x4_f32(false, a, false, b, (short)0, c,
                                               false, false);
}

__device__ __forceinline__ float fetchH(const float* xb, const float* fb, int c) {
  // gathered input channel: 0..2 = xyz, 3..66 = features, >=67 = zero pad
  if (c < 3)  return xb[c * NPTS];
  if (c < C1) return fb[(c - 3) * NPTS];
  return 0.f;
}

// ---------------------------------------------------------------- ball query
// One wave32 per query point; scan N in index order, keep first 32 hits.
__global__ void ball_query_kernel(const float* __restrict__ xyz,
                                  const float* __restrict__ q,
                                  int* __restrict__ idx) {
  int wave = blockIdx.x * (blockDim.x >> 5) + (threadIdx.x >> 5);
  int lane = threadIdx.x & 31;
  int b = wave >> 10;          // / SQ
  int s = wave & (SQ - 1);
  float qx = q[(b * SQ + s) * 3 + 0];
  float qy = q[(b * SQ + s) * 3 + 1];
  float qz = q[(b * SQ + s) * 3 + 2];
  const float* xb = xyz + (size_t)b * 3 * NPTS;
  int* out = idx + ((size_t)b * SQ + s) * NSAMP;
  int collected = 0;
  int first = NPTS - 1;        // JAX gather clamp if no neighbor
  for (int n0 = 0; n0 < NPTS; n0 += 32) {
    int n = n0 + lane;
    float dx = xb[n] - qx;
    float dy = xb[NPTS + n] - qy;
    float dz = xb[2 * NPTS + n] - qz;
    bool hit = (dx * dx + dy * dy + dz * dz) <= RAD2;
    unsigned m = (unsigned)__ballot(hit);
    if (m) {
      if (collected == 0) first = n0 + (__ffs(m) - 1);
      if (hit) {
        int slot = collected + __popc(m & ((1u << lane) - 1u));
        if (slot < NSAMP) out[slot] = n;
      }
      collected += __popc(m);
      if (collected >= NSAMP) break;
    }
  }
  if (collected < NSAMP) {
    for (int t = collected + lane; t < NSAMP; t += 32) out[t] = first;
  }
}

// --------------------------------------------------- layer 1: gather + GEMM
// Two 16x16 output tiles per wave (32 columns) sharing one A tile per k-step.
__global__ void gemm1_kernel(const float* __restrict__ xyz,
                             const float* __restrict__ pts,
                             const int* __restrict__ idx,
                             const float* __restrict__ W1,
                             const float* __restrict__ b1,
                             float* __restrict__ y1) {
  int wave = blockIdx.x * (blockDim.x >> 5) + (threadIdx.x >> 5);
  int lane = threadIdx.x & 31;
  int half = lane >> 4, l15 = lane & 15;
  int otile = (wave & 3) * 16;
  int ptile = (wave >> 2) * 32;
  int arow  = otile + l15;

  int p0 = ptile + l15, p1 = p0 + 16;
  int b0 = p0 >> 15, ks0 = (p0 >> 10) & 31, s0 = p0 & (SQ - 1);
  int b1_ = p1 >> 15, ks1 = (p1 >> 10) & 31, s1 = p1 & (SQ - 1);
  int j0 = idx[(((size_t)b0 << 10) + s0) * NSAMP + ks0];
  int j1 = idx[(((size_t)b1_ << 10) + s1) * NSAMP + ks1];
  const float* xb0 = xyz + (size_t)b0 * 3 * NPTS + j0;
  const float* fb0 = pts + (size_t)b0 * 64 * NPTS + j0;
  const float* xb1 = xyz + (size_t)b1_ * 3 * NPTS + j1;
  const float* fb1 = pts + (size_t)b1_ * 64 * NPTS + j1;

  v8f acc0 = {0.f, 0.f, 0.f, 0.f, 0.f, 0.f, 0.f, 0.f};
  v8f acc1 = {0.f, 0.f, 0.f, 0.f, 0.f, 0.f, 0.f, 0.f};
  for (int kk = 0; kk < 17; ++kk) {   // K padded 67 -> 68
    int c0 = kk * 4 + half * 2;
    v2f a, bb0, bb1;
    a.x = (c0     < C1) ? W1[arow * C1 + c0]     : 0.f;
    a.y = (c0 + 1 < C1) ? W1[arow * C1 + c0 + 1] : 0.f;
    bb0.x = fetchH(xb0, fb0, c0);
    bb0.y = fetchH(xb0, fb0, c0 + 1);
    bb1.x = fetchH(xb1, fb1, c0);
    bb1.y = fetchH(xb1, fb1, c0 + 1);
    acc0 = wmma4(a, bb0, acc0);
    acc1 = wmma4(a, bb1, acc1);
  }
#pragma unroll
  for (int r = 0; r < 8; ++r) {
    int o = otile + r + half * 8;
    float bo = b1[o];
    y1[(size_t)o * PTOT + p0] = acc0[r] + bo;
    y1[(size_t)o * PTOT + p1] = acc1[r] + bo;
  }
}

// ---------------------- per-channel mean/var -> BN affine (scale, shift)
__global__ void stats_kernel(const float* __restrict__ y,
                             const float* __restrict__ g,
                             const float* __restrict__ be,
                             float* __restrict__ aff) {
  __shared__ float ssum[256], ssq[256];
  int c = blockIdx.x, tid = threadIdx.x;
  const float* row = y + (size_t)c * PTOT;
  float s = 0.f, s2 = 0.f;
  for (int i = tid; i < PTOT; i += blockDim.x) {
    float v = row[i]; s += v; s2 += v * v;
  }
  ssum[tid] = s; ssq[tid] = s2; __syncthreads();
  for (int st = 128; st > 0; st >>= 1) {
    if (tid < st) { ssum[tid] += ssum[tid + st]; ssq[tid] += ssq[tid + st]; }
    __syncthreads();
  }
  if (tid == 0) {
    float inv = 1.0f / (float)PTOT;
    float mean = ssum[0] * inv;
    float var  = ssq[0] * inv - mean * mean;
    float sc   = g[c] * rsqrtf(var + EPS_BN);
    aff[2 * c] = sc;
    aff[2 * c + 1] = be[c] - mean * sc;
  }
}

// ----------------------------- layer 2: y2 = W2 * relu(affine1(y1)) + b2
__global__ void gemm2_kernel(const float* __restrict__ y1,
                             const float* __restrict__ aff1,
                             const float* __restrict__ W2,
                             const float* __restrict__ b2,
                             float* __restrict__ y2) {
  int wave = blockIdx.x * (blockDim.x >> 5) + (threadIdx.x >> 5);
  int lane = threadIdx.x & 31;
  int half = lane >> 4, l15 = lane & 15;
  int otile = (wave & 3) * 16;
  int p0 = (wave >> 2) * 32 + l15;
  int p1 = p0 + 16;
  int arow = otile + l15;
  v8f acc0 = {0.f, 0.f, 0.f, 0.f, 0.f, 0.f, 0.f, 0.f};
  v8f acc1 = {0.f, 0.f, 0.f, 0.f, 0.f, 0.f, 0.f, 0.f};
  for (int kk = 0; kk < 16; ++kk) {   // C = 64
    int c0 = kk * 4 + half * 2;
    float sc0 = aff1[2 * c0],     sh0 = aff1[2 * c0 + 1];
    float sc1 = aff1[2 * c0 + 2], sh1 = aff1[2 * c0 + 3];
    v2f a, bb0, bb1;
    a.x = W2[arow * 64 + c0];
    a.y = W2[arow * 64 + c0 + 1];
    bb0.x = fmaxf(y1[(size_t)c0 * PTOT + p0]       * sc0 + sh0, 0.f);
    bb0.y = fmaxf(y1[(size_t)(c0 + 1) * PTOT + p0] * sc1 + sh1, 0.f);
    bb1.x = fmaxf(y1[(size_t)c0 * PTOT + p1]       * sc0 + sh0, 0.f);
    bb1.y = fmaxf(y1[(size_t)(c0 + 1) * PTOT + p1] * sc1 + sh1, 0.f);
    acc0 = wmma4(a, bb0, acc0);
    acc1 = wmma4(a, bb1, acc1);
  }
#pragma unroll
  for (int r = 0; r < 8; ++r) {
    int o = otile + r + half * 8;
    float bo = b2[o];
    y2[(size_t)o * PTOT + p0] = acc0[r] + bo;
    y2[(size_t)o * PTOT + p1] = acc1[r] + bo;
  }
}

// -------- layer 3 fused with k-reduction: per wave = (otile, b, s-tile),
// loop over all 32 samples; A tile, BN affine and bias hoisted to registers.
__global__ void gemm3_kernel(const float* __restrict__ y2,
                             const float* __restrict__ aff2,
                             const float* __restrict__ W3,
                             const float* __restrict__ b3,
                             float* __restrict__ ymax, float* __restrict__ ymin,
                             float* __restrict__ ysum, float* __restrict__ ysq) {
  int wave = blockIdx.x * (blockDim.x >> 5) + (threadIdx.x >> 5);
  int lane = threadIdx.x & 31;
  int half = lane >> 4, l15 = lane & 15;
  int otile = wave >> 8;           // 0..7
  int b     = (wave >> 6) & 3;
  int stile = wave & 63;
  int arow  = otile * 16 + l15;
  int scol  = stile * 16 + l15;

  v2f aW[16], scv[16], shv[16];
#pragma unroll
  for (int kk = 0; kk < 16; ++kk) {
    int c0 = kk * 4 + half * 2;
    aW[kk].x  = W3[arow * 64 + c0];
    aW[kk].y  = W3[arow * 64 + c0 + 1];
    scv[kk].x = aff2[2 * c0];     shv[kk].x = aff2[2 * c0 + 1];
    scv[kk].y = aff2[2 * c0 + 2]; shv[kk].y = aff2[2 * c0 + 3];
  }
  float bias[8];
#pragma unroll
  for (int r = 0; r < 8; ++r) bias[r] = b3[otile * 16 + r + half * 8];

  v8f mx, mn, sm, s2;
#pragma unroll
  for (int r = 0; r < 8; ++r) { mx[r] = -FLT_MAX; mn[r] = FLT_MAX; sm[r] = 0.f; s2[r] = 0.f; }

  for (int k = 0; k < NSAMP; ++k) {
    const float* bp = y2 + (size_t)(b * (NSAMP * SQ) + k * SQ + scol);
    v8f acc = {0.f, 0.f, 0.f, 0.f, 0.f, 0.f, 0.f, 0.f};
#pragma unroll
    for (int kk = 0; kk < 16; ++kk) {
      int c0 = kk * 4 + half * 2;
      v2f bb;
      bb.x = fmaxf(bp[(size_t)c0 * PTOT]       * scv[kk].x + shv[kk].x, 0.f);
      bb.y = fmaxf(bp[(size_t)(c0 + 1) * PTOT] * scv[kk].y + shv[kk].y, 0.f);
      acc = wmma4(aW[kk], bb, acc);
    }
#pragma unroll
    for (int r = 0; r < 8; ++r) {
      float v = acc[r] + bias[r];
      mx[r] = fmaxf(mx[r], v); mn[r] = fminf(mn[r], v);
      sm[r] += v; s2[r] += v * v;
    }
  }
  int col = b * SQ + scol;
#pragma unroll
  for (int r = 0; r < 8; ++r) {
    size_t off = (size_t)(otile * 16 + r + half * 8) * COLS3 + col;
    ymax[off] = mx[r]; ymin[off] = mn[r]; ysum[off] = sm[r]; ysq[off] = s2[r];
  }
}

// ------------- layer-3 stats from the per-(b,s) partial sums over k
__global__ void stats3_kernel(const float* __restrict__ ysum,
                              const float* __restrict__ ysq,
                              const float* __restrict__ g,
                              const float* __restrict__ be,
                              float* __restrict__ aff) {
  __shared__ float ssum[256], ssq[256];
  int c = blockIdx.x, tid = threadIdx.x;
  float s = 0.f, s2 = 0.f;
  for (int i = tid; i < COLS3; i += blockDim.x) {
    s  += ysum[(size_t)c * COLS3 + i];
    s2 += ysq[(size_t)c * COLS3 + i];
  }
  ssum[tid] = s; ssq[tid] = s2; __syncthreads();
  for (int st = 128; st > 0; st >>= 1) {
    if (tid < st) { ssum[tid] += ssum[tid + st]; ssq[tid] += ssq[tid + st]; }
    __syncthreads();
  }
  if (tid == 0) {
    float inv = 1.0f / (float)PTOT;      // count = B*NSAMP*S
    float mean = ssum[0] * inv;
    float var  = ssq[0] * inv - mean * mean;
    float sc   = g[c] * rsqrtf(var + EPS_BN);
    aff[2 * c] = sc;
    aff[2 * c + 1] = be[c] - mean * sc;
  }
}

// ---- max_k relu(affine3(y3)) : affine with sc>=0 commutes with max, else min
__global__ void finalize_kernel(const float* __restrict__ ymax,
                                const float* __restrict__ ymin,
                                const float* __restrict__ aff,
                                float* __restrict__ out) {
  int i = blockIdx.x * blockDim.x + threadIdx.x;   // 0 .. O3*COLS3-1
  int o = i >> 12;                                 // / COLS3
  int col = i & (COLS3 - 1);
  float sc = aff[2 * o], sh = aff[2 * o + 1];
  float v = (sc >= 0.f) ? ymax[i] : ymin[i];
  float r = fmaxf(v * sc + sh, 0.f);
  int b = col >> 10, s = col & (SQ - 1);
  out[QOFF + ((size_t)b * O3 + o) * SQ + s] = r;
}

extern "C" void kernel_launch(void* const* d_in, const int* in_sizes, int n_in,
                              void* d_out, int out_size, void* d_ws, size_t ws_size,
                              hipStream_t stream) {
  const float* xyz = (const float*)d_in[0];
  const float* pts = (const float*)d_in[1];
  const float* qp  = (const float*)d_in[2];
  const float* W1  = (const float*)d_in[3];
  const float* b1  = (const float*)d_in[4];
  const float* g1  = (const float*)d_in[5];
  const float* be1 = (const float*)d_in[6];
  const float* W2  = (const float*)d_in[7];
  const float* b2  = (const float*)d_in[8];
  const float* g2  = (const float*)d_in[9];
  const float* be2 = (const float*)d_in[10];
  const float* W3  = (const float*)d_in[11];
  const float* b3  = (const float*)d_in[12];
  const float* g3  = (const float*)d_in[13];
  const float* be3 = (const float*)d_in[14];
  float* out = (float*)d_out;

  float* ws    = (float*)d_ws;
  int*   idx   = (int*)ws;                       // PTOT ints
  float* y1    = ws + PTOT;
  float* y2    = y1 + (size_t)O1 * PTOT;
  float* ysum  = y2 + (size_t)O2 * PTOT;
  float* ysq   = ysum + (size_t)O3 * COLS3;
  float* ymax  = ysq  + (size_t)O3 * COLS3;
  float* ymin  = ymax + (size_t)O3 * COLS3;
  float* aff1  = ymin + (size_t)O3 * COLS3;
  float* aff2  = aff1 + 2 * O1;
  float* aff3  = aff2 + 2 * O2;

  hipMemcpyAsync(out, qp, QOFF * sizeof(float), hipMemcpyDeviceToDevice, stream);

  ball_query_kernel<<<(BATCH * SQ) / 8, 256, 0, stream>>>(xyz, qp, idx);
  gemm1_kernel<<<(4 * (PTOT / 32)) / 8, 256, 0, stream>>>(xyz, pts, idx, W1, b1, y1);
  stats_kernel<<<O1, 256, 0, stream>>>(y1, g1, be1, aff1);
  gemm2_kernel<<<(4 * (PTOT / 32)) / 8, 256, 0, stream>>>(y1, aff1, W2, b2, y2);
  stats_kernel<<<O2, 256, 0, stream>>>(y2, g2, be2, aff2);
  gemm3_kernel<<<(8 * (COLS3 / 16)) / 8, 256, 0, stream>>>(y2, aff2, W3, b3,
                                                           ymax, ymin, ysum, ysq);
  stats3_kernel<<<O3, 256, 0, stream>>>(ysum, ysq, g3, be3, aff3);
  finalize_kernel<<<(O3 * COLS3) / 256, 256, 0, stream>>>(ymax, ymin, aff3, out);
}